// GraphConv_3032246911605
// MI455X (gfx1250) — compile-verified
//
#include <hip/hip_runtime.h>

typedef __attribute__((ext_vector_type(2))) float v2f;
typedef __attribute__((ext_vector_type(8))) float v8f;

#define N_NODES 100000

// ---------------------------------------------------------------------------
// Degree / normalization
// ---------------------------------------------------------------------------
__global__ void deg_init(float* deg, int n) {
    int i = blockIdx.x * blockDim.x + threadIdx.x;
    if (i < n) deg[i] = 1.0f;  // self-loop contributes 1
}

__global__ void deg_count(const long long* __restrict__ dst, float* deg, int E) {
    int e = blockIdx.x * blockDim.x + threadIdx.x;
    if (e < E) atomicAdd(&deg[dst[e]], 1.0f);
}

__global__ void deg_rsqrt(float* deg, int n) {
    int i = blockIdx.x * blockDim.x + threadIdx.x;
    if (i < n) deg[i] = rsqrtf(deg[i]);  // deg >= 1 always
}

// ---------------------------------------------------------------------------
// Coalesced per-edge norm precompute: nrm[e] = dinv[src[e]] * dinv[dst[e]]
// ---------------------------------------------------------------------------
__global__ void edge_norm(const long long* __restrict__ src, const long long* __restrict__ dst,
                          const float* __restrict__ dinv, float* __restrict__ nrm, int E) {
    int e = blockIdx.x * blockDim.x + threadIdx.x;
    if (e < E) nrm[e] = dinv[src[e]] * dinv[dst[e]];
}

// ---------------------------------------------------------------------------
// WMMA-ready weight packing (zero-padded). For each (nTile, kQuad), store the
// two floats lane L needs for its B fragment contiguously, so the GEMM loads
// one aligned b64 per WMMA:
//   flat[( (nTile*Kpad/4 + kq)*32 + lane )*2 + j] = W[kq*4 + 2*(lane>>4)+j][nTile*16 + (lane&15)]
// ---------------------------------------------------------------------------
__global__ void pack_w_wmma(const float* __restrict__ w, float* __restrict__ bp,
                            int K, int Nin, int Kpad, int Npad) {
    int total = Kpad * Npad;  // same element count, permuted layout
    int i = blockIdx.x * blockDim.x + threadIdx.x;
    if (i >= total) return;
    int kq4 = Kpad >> 2;
    int j = i & 1;
    int lane = (i >> 1) & 31;
    int rest = i >> 6;
    int kq = rest % kq4;
    int nTile = rest / kq4;
    int k = (kq << 2) + ((lane >> 4) << 1) + j;
    int n = (nTile << 4) + (lane & 15);
    bp[i] = (k < K && n < Nin) ? w[k * Nin + n] : 0.0f;
}

// ---------------------------------------------------------------------------
// WMMA fp32 GEMM: Y[M x Npad] = X[M x Kpad] @ W[Kpad x Npad] (packed as Bp).
// One wave32 computes FOUR 16x16 M-tiles at one N-tile, sharing each B
// fragment across 4 v_wmma_f32_16x16x4_f32 issues (4x weight-stream reuse,
// 4 independent WMMAs in flight). All branches are wave-uniform so EXEC
// stays all-1s around every WMMA (ISA 7.12 requirement).
// VGPR layouts per CDNA5 ISA 7.12.2:
//   A/B vgpr j : K = j + 2*(lane>>4), row/col = lane&15
//   C/D vgpr r : row = r + 8*(lane>>4), col = lane&15
// ---------------------------------------------------------------------------
template <bool FUSE_BIAS_RELU>
__global__ void wmma_gemm(const float* __restrict__ X, const float* __restrict__ Bp,
                          const float* __restrict__ bias, float* __restrict__ Y,
                          int M, int Kpad, int Npad, int nOut) {
    const int nTiles = Npad >> 4;
    const int tilesM = M >> 4;                 // M divisible by 16
    const int mGroups = (tilesM + 3) >> 2;     // 4 M-tiles per wave
    const int waveId = blockIdx.x * (blockDim.x >> 5) + (threadIdx.x >> 5);
    if (waveId >= mGroups * nTiles) return;    // wave-uniform exit

    const int nTile = waveId % nTiles;
    const int mt0 = (waveId / nTiles) << 2;
    const int lane = threadIdx.x & 31;
    const int half = lane >> 4;                // 0: lanes 0-15, 1: lanes 16-31
    const int idx = lane & 15;
    const int col = (nTile << 4) + idx;
    const int kq4 = Kpad >> 2;

    // Lane's B stream: one v2f (b64) per k-quad.
    const v2f* __restrict__ bpp = (const v2f*)Bp + (size_t)nTile * kq4 * 32 + lane;

    // Per-tile A row pointers (clamped for the ragged last group; loads stay
    // in-bounds, results discarded at store time).
    const float* xr[4];
#pragma unroll
    for (int t = 0; t < 4; ++t) {
        int mTile = mt0 + t;
        if (mTile >= tilesM) mTile = tilesM - 1;
        xr[t] = X + (size_t)((mTile << 4) + idx) * Kpad + (half << 1);
    }

    v8f acc[4] = {{}, {}, {}, {}};
    for (int k0 = 0; k0 < Kpad; k0 += 4) {
        v2f b = bpp[(k0 >> 2) * 32];
#pragma unroll
        for (int t = 0; t < 4; ++t) {
            v2f a = *(const v2f*)(xr[t] + k0);  // 8B-aligned: row stride %16==0, k0+2*half even
            acc[t] = __builtin_amdgcn_wmma_f32_16x16x4_f32(
                         /*neg_a=*/false, a, /*neg_b=*/false, b,
                         /*c_mod=*/(short)0, acc[t], /*reuse_a=*/false, /*reuse_b=*/false);
        }
    }

#pragma unroll
    for (int t = 0; t < 4; ++t) {
        int mTile = mt0 + t;
        if (mTile >= tilesM) break;            // wave-uniform
        int mBase = (mTile << 4) + (half << 3);
#pragma unroll
        for (int r = 0; r < 8; ++r) {
            float v = acc[t][r];
            if (FUSE_BIAS_RELU) {
                v += (col < nOut) ? bias[col] : 0.0f;
                v = fmaxf(v, 0.0f);
            }
            Y[(size_t)(mBase + r) * Npad + col] = v;
        }
    }
}

// ---------------------------------------------------------------------------
// Self-loop term (also initializes Y, incl. zeroing pad columns since xW pad=0):
//   y[i,f] = xw[i,f] * dinv[i]^2
// ---------------------------------------------------------------------------
__global__ void self_loop_init(const float* __restrict__ xw, const float* __restrict__ dinv,
                               float* __restrict__ y, int n_elems, int stride) {
    int i = blockIdx.x * blockDim.x + threadIdx.x;
    if (i >= n_elems) return;
    float dv = dinv[i / stride];
    y[i] = xw[i] * dv * dv;
}

// ---------------------------------------------------------------------------
// Edge aggregation: one wave per edge, lanes stride the feature dimension.
//   y[dst,f] += xw[src,f] * nrm[e]
// L2-resident gather + global_atomic_add_f32 scatter (L2 atomic units).
// ---------------------------------------------------------------------------
__global__ void aggregate(const float* __restrict__ xw,
                          const long long* __restrict__ src,
                          const long long* __restrict__ dst,
                          const float* __restrict__ nrm,
                          float* __restrict__ y, int E, int stride, int dim) {
    int gw = (blockIdx.x * blockDim.x + threadIdx.x) >> 5;
    int lane = threadIdx.x & 31;
    if (gw >= E) return;
    long long s = src[gw];
    long long d = dst[gw];
    float norm = nrm[gw];
    const float* __restrict__ xs = xw + (size_t)s * stride;
    float* __restrict__ yd = y + (size_t)d * stride;
    for (int f = lane; f < dim; f += 32)
        atomicAdd(yd + f, xs[f] * norm);
}

// ---------------------------------------------------------------------------
// Bias + ReLU (pad columns: 0 + 0 -> 0, stays zero)
// ---------------------------------------------------------------------------
__global__ void bias_relu(float* __restrict__ y, const float* __restrict__ b,
                          int n_elems, int stride, int dim) {
    int i = blockIdx.x * blockDim.x + threadIdx.x;
    if (i >= n_elems) return;
    int f = i % stride;
    y[i] = fmaxf(y[i] + (f < dim ? b[f] : 0.0f), 0.0f);
}

// ---------------------------------------------------------------------------
// Final 10 -> 1 dense layer + ReLU
// ---------------------------------------------------------------------------
__global__ void final_dense(const float* __restrict__ x, const float* __restrict__ w,
                            const float* __restrict__ b, float* __restrict__ out,
                            int N, int stride) {
    int i = blockIdx.x * blockDim.x + threadIdx.x;
    if (i >= N) return;
    const float* xr = x + (size_t)i * stride;
    float s = b[0];
#pragma unroll
    for (int f = 0; f < 10; ++f) s += xr[f] * w[f];
    out[i] = fmaxf(s, 0.0f);
}

// ---------------------------------------------------------------------------
extern "C" void kernel_launch(void* const* d_in, const int* in_sizes, int n_in,
                              void* d_out, int out_size, void* d_ws, size_t ws_size,
                              hipStream_t stream) {
    const float* X0  = (const float*)d_in[0];
    const float* W1  = (const float*)d_in[1];  const float* b1  = (const float*)d_in[2];
    const float* W2  = (const float*)d_in[3];  const float* b2  = (const float*)d_in[4];
    const float* W3  = (const float*)d_in[5];  const float* b3  = (const float*)d_in[6];
    const float* Wl1 = (const float*)d_in[7];  const float* bl1 = (const float*)d_in[8];
    const float* Wl2 = (const float*)d_in[9];  const float* bl2 = (const float*)d_in[10];
    const float* Wl3 = (const float*)d_in[11]; const float* bl3 = (const float*)d_in[12];
    const long long* ei = (const long long*)d_in[13];

    const int N = N_NODES;
    const int E = in_sizes[13] / 2;
    const long long* src = ei;
    const long long* dstp = ei + E;

    // Workspace (floats): dinv | edge norms | buf0 (N*112) | buf1 (N*112) | packed weights
    float* ws   = (float*)d_ws;
    float* dinv = ws;
    float* nrm  = dinv + N;
    float* buf0 = nrm + E;
    float* buf1 = buf0 + (size_t)N * 112;
    float* wp1  = buf1 + (size_t)N * 112;   // 64  x 112
    float* wp2  = wp1 + 64 * 112;           // 112 x 64
    float* wp3  = wp2 + 112 * 64;           // 64  x 32
    float* wpl1 = wp3 + 64 * 32;            // 32  x 32
    float* wpl2 = wpl1 + 32 * 32;           // 32  x 16

    const int T = 256;
    auto blocks = [](long long n, int t) { return (int)((n + t - 1) / t); };
    auto gemm_blocks = [&](int M, int Npad) {
        int waves = (((M >> 4) + 3) >> 2) * (Npad >> 4);  // 4 M-tiles per wave
        return (waves + (T >> 5) - 1) / (T >> 5);
    };

    // --- D^-1/2 and per-edge norms ---
    deg_init <<<blocks(N, T), T, 0, stream>>>(dinv, N);
    deg_count<<<blocks(E, T), T, 0, stream>>>(dstp, dinv, E);
    deg_rsqrt<<<blocks(N, T), T, 0, stream>>>(dinv, N);
    edge_norm<<<blocks(E, T), T, 0, stream>>>(src, dstp, dinv, nrm, E);

    // --- pack weights into WMMA-ready zero-padded layout ---
    pack_w_wmma<<<blocks(64 * 112, T), T, 0, stream>>>(W1,  wp1,  64, 100,  64, 112);
    pack_w_wmma<<<blocks(112 * 64, T), T, 0, stream>>>(W2,  wp2, 100,  50, 112,  64);
    pack_w_wmma<<<blocks(64 * 32,  T), T, 0, stream>>>(W3,  wp3,  50,  25,  64,  32);
    pack_w_wmma<<<blocks(32 * 32,  T), T, 0, stream>>>(Wl1, wpl1, 25,  25,  32,  32);
    pack_w_wmma<<<blocks(32 * 16,  T), T, 0, stream>>>(Wl2, wpl2, 25,  10,  32,  16);

    // --- GCN layer 1: 64 -> 100 (pad 112) ---
    wmma_gemm<false><<<gemm_blocks(N, 112), T, 0, stream>>>(X0, wp1, nullptr, buf0, N, 64, 112, 100);
    self_loop_init<<<blocks((long long)N * 112, T), T, 0, stream>>>(buf0, dinv, buf1, N * 112, 112);
    aggregate<<<blocks((long long)E * 32, T), T, 0, stream>>>(buf0, src, dstp, nrm, buf1, E, 112, 100);
    bias_relu<<<blocks((long long)N * 112, T), T, 0, stream>>>(buf1, b1, N * 112, 112, 100);

    // --- GCN layer 2: 100 (stride 112) -> 50 (pad 64) ---
    wmma_gemm<false><<<gemm_blocks(N, 64), T, 0, stream>>>(buf1, wp2, nullptr, buf0, N, 112, 64, 50);
    self_loop_init<<<blocks((long long)N * 64, T), T, 0, stream>>>(buf0, dinv, buf1, N * 64, 64);
    aggregate<<<blocks((long long)E * 32, T), T, 0, stream>>>(buf0, src, dstp, nrm, buf1, E, 64, 50);
    bias_relu<<<blocks((long long)N * 64, T), T, 0, stream>>>(buf1, b2, N * 64, 64, 50);

    // --- GCN layer 3: 50 (stride 64) -> 25 (pad 32) ---
    wmma_gemm<false><<<gemm_blocks(N, 32), T, 0, stream>>>(buf1, wp3, nullptr, buf0, N, 64, 32, 25);
    self_loop_init<<<blocks((long long)N * 32, T), T, 0, stream>>>(buf0, dinv, buf1, N * 32, 32);
    aggregate<<<blocks((long long)E * 32, T), T, 0, stream>>>(buf0, src, dstp, nrm, buf1, E, 32, 25);
    bias_relu<<<blocks((long long)N * 32, T), T, 0, stream>>>(buf1, b3, N * 32, 32, 25);

    // --- Dense 25 -> 25 (pad 32), fused bias+ReLU ---
    wmma_gemm<true><<<gemm_blocks(N, 32), T, 0, stream>>>(buf1, wpl1, bl1, buf0, N, 32, 32, 25);
    // --- Dense 25 -> 10 (pad 16), fused bias+ReLU ---
    wmma_gemm<true><<<gemm_blocks(N, 16), T, 0, stream>>>(buf0, wpl2, bl2, buf1, N, 32, 16, 10);
    // --- Dense 10 -> 1 + ReLU ---
    final_dense<<<blocks(N, T), T, 0, stream>>>(buf1, Wl3, bl3, (float*)d_out, N, 16);
}